// AtomicLinear_78529182040839
// MI455X (gfx1250) — compile-verified
//
#include <hip/hip_runtime.h>
#include <hip/hip_bf16.h>

// Problem: B=4096 (M), IN=128 (K), OUT=512 (N), all f32.
// out = x @ (weight * softmax_rowsum)^T + bias

typedef __attribute__((ext_vector_type(2))) float v2f;
typedef __attribute__((ext_vector_type(8))) float v8f;
typedef __attribute__((ext_vector_type(4))) int   v4i_;

#define M_DIM 4096
#define K_DIM 128
#define N_DIM 512

#define M_TILE 128
#define N_TILE 64
#define LDSS   132   // padded LDS stride (floats): bank step 132%64=4 -> conflict-free frag reads

// CDNA5 async global->LDS path (direct memory->LDS DMA, tracked by ASYNCcnt).
#if __has_builtin(__builtin_amdgcn_global_load_async_to_lds_b128)
#define HAVE_ASYNC_LDS 1
#else
#define HAVE_ASYNC_LDS 0
#endif

// Builtin signature (from hipcc diagnostic): param0 = int4 addrspace(1)*,
// param1 = int4 addrspace(3)*, then imm offset, imm cpol.
#define GLOBAL_AS_V4I(p) ((__attribute__((address_space(1))) v4i_*)(p))
#define LDS_AS_V4I(p)    ((__attribute__((address_space(3))) v4i_*)(p))

__device__ __forceinline__ void wait_asynccnt0() {
#if __has_builtin(__builtin_amdgcn_s_wait_asynccnt)
  __builtin_amdgcn_s_wait_asynccnt(0);
#else
  asm volatile("s_wait_asynccnt 0x0" ::: "memory");
#endif
}

// ---------------------------------------------------------------------------
// Kernel 1: per-row softmax row-sum of (logits + gumbel)/T, T=1.
// grid = 128 blocks (one per row), block = 128 threads.
// ---------------------------------------------------------------------------
__global__ __launch_bounds__(128) void softmax_rowsum_kernel(
    const float* __restrict__ logits, const float* __restrict__ gumbel,
    float* __restrict__ rowsum) {
  __shared__ float red[128];
  const int r = blockIdx.x;
  const int t = threadIdx.x;

  const float z = logits[r * K_DIM + t] + gumbel[r * K_DIM + t];

  red[t] = z;
  __syncthreads();
#pragma unroll
  for (int s = 64; s > 0; s >>= 1) {
    if (t < s) red[t] = fmaxf(red[t], red[t + s]);
    __syncthreads();
  }
  const float m = red[0];
  __syncthreads();

  const float e = __expf(z - m);
  red[t] = e;
  __syncthreads();
#pragma unroll
  for (int s = 64; s > 0; s >>= 1) {
    if (t < s) red[t] += red[t + s];
    __syncthreads();
  }
  const float S = red[0];
  __syncthreads();

  const float p = e / S;
  red[t] = p;
  __syncthreads();
#pragma unroll
  for (int s = 64; s > 0; s >>= 1) {
    if (t < s) red[t] += red[t + s];
    __syncthreads();
  }
  if (t == 0) rowsum[r] = red[0];
}

// ---------------------------------------------------------------------------
// Kernel 2: w_scaled[o,i] = weight[o,i] * rowsum[i].  256 KB, L2-resident.
// 16384 float4 / (64 blocks * 256 threads) = 1 float4 per thread.
// ---------------------------------------------------------------------------
__global__ __launch_bounds__(256) void scale_weight_kernel(
    const float* __restrict__ w, const float* __restrict__ rowsum,
    float* __restrict__ wscaled) {
  const int t = blockIdx.x * 256 + threadIdx.x;
  const int off = t * 4;
  float4 v = *(const float4*)(w + off);
  const float4 rs = *(const float4*)(rowsum + (off & (K_DIM - 1)));
  v.x *= rs.x; v.y *= rs.y; v.z *= rs.z; v.w *= rs.w;
  *(float4*)(wscaled + off) = v;
}

// ---------------------------------------------------------------------------
// Kernel 3: GEMM  out = x @ w_scaled^T + bias  via V_WMMA_F32_16X16X4_F32.
// block = 256 threads (8 wave32), tile = 128x64, grid = (32, 8).
// Tiles staged via GLOBAL_LOAD_ASYNC_TO_LDS_B128 when available.
// ---------------------------------------------------------------------------
__global__ __launch_bounds__(256) void fused_wmma_gemm_kernel(
    const float* __restrict__ x, const float* __restrict__ wscaled,
    const float* __restrict__ bias, float* __restrict__ out) {
  __shared__ float sA[M_TILE * LDSS];  // x tile, [m][k], padded
  __shared__ float sB[N_TILE * LDSS];  // scaled weight tile, [n][k], padded

  const int tid  = threadIdx.x;
  const int lane = tid & 31;
  const int wave = tid >> 5;
  const int m0 = blockIdx.x * M_TILE;
  const int n0 = blockIdx.y * N_TILE;

  // ---- stage tiles into LDS ------------------------------------------
  {
    const int col4  = (tid & 31) * 4;  // 32 float4 cover K=128
    const int rbase = tid >> 5;        // 8 rows per pass

#if HAVE_ASYNC_LDS
#pragma unroll
    for (int r = rbase; r < M_TILE; r += 8) {
      __builtin_amdgcn_global_load_async_to_lds_b128(
          GLOBAL_AS_V4I(x + (size_t)(m0 + r) * K_DIM + col4),
          LDS_AS_V4I(&sA[r * LDSS + col4]), 0, 0);
    }
#pragma unroll
    for (int r = rbase; r < N_TILE; r += 8) {
      __builtin_amdgcn_global_load_async_to_lds_b128(
          GLOBAL_AS_V4I(wscaled + (size_t)(n0 + r) * K_DIM + col4),
          LDS_AS_V4I(&sB[r * LDSS + col4]), 0, 0);
    }
    wait_asynccnt0();  // each wave waits for its own async copies
#else
#pragma unroll
    for (int r = rbase; r < M_TILE; r += 8) {
      const float4 v = *(const float4*)(x + (size_t)(m0 + r) * K_DIM + col4);
      *(float4*)(&sA[r * LDSS + col4]) = v;
    }
#pragma unroll
    for (int r = rbase; r < N_TILE; r += 8) {
      const float4 v = *(const float4*)(wscaled + (size_t)(n0 + r) * K_DIM + col4);
      *(float4*)(&sB[r * LDSS + col4]) = v;
    }
#endif
  }
  __syncthreads();  // cross-wave visibility of LDS tiles

  // ---- wave tiling: 4 waves along M, 2 along N; 32x32 per wave ---------
  const int wm    = (wave & 3) * 32;
  const int wn    = (wave >> 2) * 32;
  const int lmod  = lane & 15;
  const int koff  = (lane >> 4) * 2;  // ISA A/B frag layout: k0 = 2*(lane/16)

  v8f acc00 = {}, acc01 = {}, acc10 = {}, acc11 = {};

#pragma unroll
  for (int k = 0; k < K_DIM; k += 4) {
    const v2f a0 = *(const v2f*)(&sA[(wm + lmod)      * LDSS + k + koff]);
    const v2f a1 = *(const v2f*)(&sA[(wm + 16 + lmod) * LDSS + k + koff]);
    const v2f b0 = *(const v2f*)(&sB[(wn + lmod)      * LDSS + k + koff]);
    const v2f b1 = *(const v2f*)(&sB[(wn + 16 + lmod) * LDSS + k + koff]);

    acc00 = __builtin_amdgcn_wmma_f32_16x16x4_f32(false, a0, false, b0,
                                                  (short)0, acc00, false, false);
    acc01 = __builtin_amdgcn_wmma_f32_16x16x4_f32(false, a0, false, b1,
                                                  (short)0, acc01, false, false);
    acc10 = __builtin_amdgcn_wmma_f32_16x16x4_f32(false, a1, false, b0,
                                                  (short)0, acc10, false, false);
    acc11 = __builtin_amdgcn_wmma_f32_16x16x4_f32(false, a1, false, b1,
                                                  (short)0, acc11, false, false);
  }

  // ---- epilogue: C layout M = v + 8*(lane/16), N = lane%16; add bias ----
  const int lhalf = lane >> 4;
  const v8f accs[2][2] = {{acc00, acc01}, {acc10, acc11}};
#pragma unroll
  for (int i = 0; i < 2; ++i) {
#pragma unroll
    for (int j = 0; j < 2; ++j) {
      const int n  = n0 + wn + j * 16 + lmod;
      const float bv = bias[n];
      const int mb = m0 + wm + i * 16 + 8 * lhalf;
#pragma unroll
      for (int v = 0; v < 8; ++v) {
        out[(size_t)(mb + v) * N_DIM + n] = accs[i][j][v] + bv;
      }
    }
  }
}

// ---------------------------------------------------------------------------
extern "C" void kernel_launch(void* const* d_in, const int* in_sizes, int n_in,
                              void* d_out, int out_size, void* d_ws, size_t ws_size,
                              hipStream_t stream) {
  const float* x      = (const float*)d_in[0];  // [4096,128]
  const float* weight = (const float*)d_in[1];  // [512,128]
  const float* bias   = (const float*)d_in[2];  // [512]
  const float* logits = (const float*)d_in[3];  // [128,128]
  const float* gumbel = (const float*)d_in[4];  // [128,128]
  float* out     = (float*)d_out;               // [4096,512]
  float* rowsum  = (float*)d_ws;                // [128] scratch
  float* wscaled = (float*)((char*)d_ws + 512); // [512,128] scratch (16B aligned)

  softmax_rowsum_kernel<<<dim3(K_DIM), dim3(128), 0, stream>>>(logits, gumbel, rowsum);

  scale_weight_kernel<<<dim3((N_DIM * K_DIM) / (4 * 256)), dim3(256), 0, stream>>>(
      weight, rowsum, wscaled);

  fused_wmma_gemm_kernel<<<dim3(M_DIM / M_TILE, N_DIM / N_TILE), dim3(256), 0, stream>>>(
      x, wscaled, bias, out);
}